// HierarchicalVAE_86268713108335
// MI455X (gfx1250) — compile-verified
//
#include <hip/hip_runtime.h>
#include <hip/hip_bf16.h>
#include <math.h>

typedef __attribute__((ext_vector_type(2))) float v2f;
typedef __attribute__((ext_vector_type(8))) float v8f;

#define B_SZ     128
#define SCALE_C  12.0f   // CANVAS/2 - MARGIN
#define HALF_C   14.0f   // CANVAS/2

// ---- d_out layout (floats), tuple concatenated flat in return order ----
#define R_REND 0         // [128,1,28,28] = 100352
#define R_MU   100352    // [128,64]
#define R_LV   108544    // [128,64]
#define R_CP   116736    // [128,2,4,4,2] = 8192
#define R_WD   124928    // [128,2]
#define R_AL   125184    // [128,2]

// ---- workspace layout (floats) ----
#define WS_H1   0        // [128,256]
#define WS_H2   32768    // [128,256]
#define WS_HIN  65536    // [128,92] : cols 0..63 = z, cols 64..91 = pts_norm
#define WS_G1   77312    // [128,512]
#define WS_H2D  142848   // [128,512]
#define WS_R1   208384   // [128,512]

// ---------------------------------------------------------------------------
// f32 WMMA 16x16x4 tile GEMM: acc += A[m0:m0+16, :K] * W[:K, n0:n0+16]
// A row-major with leading dim lda; W row-major [K x N] leading dim ldb.
// Column index clamped for safe loads when N is ragged (stores predicated).
// ---------------------------------------------------------------------------
__device__ inline v8f wmma_tile_f32(const float* __restrict__ A,
                                    const float* __restrict__ W,
                                    int K, int lda, int ldb,
                                    int m0, int n0, int N, v8f acc) {
  const int lane = threadIdx.x & 31;
  const int half = lane >> 4;    // 0: K={0,1}, 1: K={2,3}
  const int l    = lane & 15;    // A: row-in-tile, B: col-in-tile
  int col = n0 + l; if (col >= N) col = N - 1;   // clamp, garbage cols unstored
  const float* __restrict__ arow = A + (size_t)(m0 + l) * lda;
  for (int k = 0; k < K; k += 4) {
    v2f a, b;
    a.x = arow[k + 2 * half + 0];
    a.y = arow[k + 2 * half + 1];
    b.x = W[(size_t)(k + 2 * half + 0) * ldb + col];
    b.y = W[(size_t)(k + 2 * half + 1) * ldb + col];
    acc = __builtin_amdgcn_wmma_f32_16x16x4_f32(false, a, false, b,
                                                (short)0, acc, false, false);
  }
  return acc;
}

__device__ inline float act_apply(float v, int act) {
  if (act == 1) return v > 0.f ? v : 0.2f * v;                       // leaky_relu 0.2
  if (act == 2) {                                                    // selu
    const float sc = 1.0507009873554805f, al = 1.6732632423543772f;
    return sc * (v > 0.f ? v : al * (__expf(v) - 1.f));
  }
  if (act == 3) return tanhf(v);
  return v;
}

template <int ACT>
__global__ void gemm_kernel(const float* __restrict__ A,
                            const float* __restrict__ W,
                            const float* __restrict__ bias,
                            float* __restrict__ out,
                            int M, int N, int K, int lda, int ldb, int ldo) {
  const int wpb  = blockDim.x >> 5;
  const int wave = blockIdx.x * wpb + (threadIdx.x >> 5);
  const int ntiles = (N + 15) >> 4;
  const int mtiles = M >> 4;
  if (wave >= ntiles * mtiles) return;            // wave-uniform: EXEC stays full
  const int mt = wave / ntiles, nt = wave % ntiles;
  const int m0 = mt << 4, n0 = nt << 4;

  v8f acc = {0.f, 0.f, 0.f, 0.f, 0.f, 0.f, 0.f, 0.f};
  acc = wmma_tile_f32(A, W, K, lda, ldb, m0, n0, N, acc);

  const int lane = threadIdx.x & 31;
  const int half = lane >> 4, l = lane & 15;
  const int col  = n0 + l;
  if (col < N) {
    const float bv = bias[col];
    #pragma unroll
    for (int r = 0; r < 8; ++r) {
      const int row = m0 + r + 8 * half;
      out[(size_t)row * ldo + col] = act_apply(acc[r] + bv, ACT);
    }
  }
}

// ---------------------------------------------------------------------------
// Fused latent head: mu, logvar (two WMMA chains sharing A loads) + reparam z.
// h2:[128,256] -> mu,lv:[128,64], z scattered into h_in[:, 0:64] (ld 92).
// ---------------------------------------------------------------------------
__global__ void latent_kernel(const float* __restrict__ h2,
                              const float* __restrict__ muW, const float* __restrict__ muB,
                              const float* __restrict__ lvW, const float* __restrict__ lvB,
                              const float* __restrict__ eps,
                              float* __restrict__ muO, float* __restrict__ lvO,
                              float* __restrict__ hin) {
  const int wave = blockIdx.x * (blockDim.x >> 5) + (threadIdx.x >> 5);
  if (wave >= 32) return;                         // 8 m-tiles x 4 n-tiles
  const int mt = wave >> 2, nt = wave & 3;
  const int m0 = mt << 4, n0 = nt << 4;
  const int lane = threadIdx.x & 31;
  const int half = lane >> 4, l = lane & 15;
  const int col = n0 + l;
  const float* __restrict__ arow = h2 + (size_t)(m0 + l) * 256;

  v8f aMu = {0.f,0.f,0.f,0.f,0.f,0.f,0.f,0.f};
  v8f aLv = aMu;
  for (int k = 0; k < 256; k += 4) {
    v2f a, bm, bl;
    a.x  = arow[k + 2 * half + 0];
    a.y  = arow[k + 2 * half + 1];
    bm.x = muW[(k + 2 * half + 0) * 64 + col];
    bm.y = muW[(k + 2 * half + 1) * 64 + col];
    bl.x = lvW[(k + 2 * half + 0) * 64 + col];
    bl.y = lvW[(k + 2 * half + 1) * 64 + col];
    aMu = __builtin_amdgcn_wmma_f32_16x16x4_f32(false, a, false, bm, (short)0, aMu, false, false);
    aLv = __builtin_amdgcn_wmma_f32_16x16x4_f32(false, a, false, bl, (short)0, aLv, false, false);
  }
  const float mb = muB[col], lb = lvB[col];
  #pragma unroll
  for (int r = 0; r < 8; ++r) {
    const int row = m0 + r + 8 * half;
    const float mu = aMu[r] + mb;
    const float lv = aLv[r] + lb;
    const float z  = mu + eps[row * 64 + col] * __expf(0.5f * lv);
    muO[row * 64 + col] = mu;
    lvO[row * 64 + col] = lv;
    hin[row * 92 + col] = z;
  }
}

// ---------------------------------------------------------------------------
// widths/alphas heads: tiny (128x2 each, K=512) -> plain dot products.
// ---------------------------------------------------------------------------
__global__ void heads_kernel(const float* __restrict__ h2d,
                             const float* __restrict__ wdW, const float* __restrict__ wdB,
                             const float* __restrict__ alW, const float* __restrict__ alB,
                             float* __restrict__ wout, float* __restrict__ aout) {
  const int tid = blockIdx.x * blockDim.x + threadIdx.x;
  if (tid >= B_SZ * 4) return;
  const int b = tid >> 2, j = tid & 3;        // j: 0,1 -> widths ; 2,3 -> alphas
  const int col = j & 1;
  const float* __restrict__ W  = (j < 2) ? wdW : alW;
  const float  bb = (j < 2) ? wdB[col] : alB[col];
  const float* __restrict__ h = h2d + (size_t)b * 512;
  float s = bb;
  for (int k = 0; k < 512; ++k) s += h[k] * W[k * 2 + col];
  const float sg = 1.0f / (1.0f + __expf(-s));
  if (j < 2) wout[b * 2 + col] = sg * 2.0f + 1.0f;
  else       aout[b * 2 + col] = sg;
}

// ---------------------------------------------------------------------------
// ref2 (tanh) + points + control-point gather-scatter into d_out.
// r1:[128,512] x ref_w2:[512,52] -> points[b, p=col/26, i=(col%26)/2, d=col&1]
// control_points[b,p,s,j,:] = points[b,p,3s+j,:]
// ---------------------------------------------------------------------------
__global__ void ref2_points_kernel(const float* __restrict__ A,
                                   const float* __restrict__ W,
                                   const float* __restrict__ bias,
                                   float* __restrict__ cp_out) {
  const int wave = blockIdx.x * (blockDim.x >> 5) + (threadIdx.x >> 5);
  if (wave >= 32) return;                       // 8 m-tiles x 4 n-tiles (N=52 -> 64)
  const int mt = wave >> 2, nt = wave & 3;
  const int m0 = mt << 4, n0 = nt << 4;

  v8f acc = {0.f,0.f,0.f,0.f,0.f,0.f,0.f,0.f};
  acc = wmma_tile_f32(A, W, 512, 512, 52, m0, n0, 52, acc);

  const int lane = threadIdx.x & 31;
  const int half = lane >> 4, l = lane & 15;
  const int col  = n0 + l;
  if (col < 52) {
    const float bv  = bias[col];
    const int p   = col / 26;
    const int rem = col % 26;
    const int i   = rem >> 1;   // point index 0..12
    const int d   = rem & 1;
    #pragma unroll
    for (int r = 0; r < 8; ++r) {
      const int row = m0 + r + 8 * half;
      const float pt = tanhf(acc[r] + bv) * SCALE_C + HALF_C;
      #pragma unroll
      for (int s = 0; s < 4; ++s) {             // shared endpoints write twice
        const int j = i - 3 * s;
        if (j >= 0 && j < 4)
          cp_out[((((row * 2 + p) * 4 + s) * 4 + j) * 2 + d)] = pt;
      }
    }
  }
}

// ---------------------------------------------------------------------------
// Rasterizer: 1 block = 1 image. Bezier sample points + widths/alphas in LDS,
// each thread owns coarse pixels, 2x2 AA, 2 paths x 32-point polyline.
// ---------------------------------------------------------------------------
__global__ void render_kernel(const float* __restrict__ cp,
                              const float* __restrict__ widths,
                              const float* __restrict__ alphas,
                              float* __restrict__ rendered) {
  const int b = blockIdx.x;
  __shared__ float qx[2][32];
  __shared__ float qy[2][32];
  __shared__ float sw[2], sa[2];
  const int tid = threadIdx.x;

  if (tid < 64) {                              // evaluate cubic Bezier samples
    const int p = tid >> 5, m = tid & 31;
    const int s = m >> 3,  t = m & 7;
    const float tt = (float)t / 7.0f, mt = 1.0f - tt;
    const float b0 = mt * mt * mt;
    const float b1 = 3.f * mt * mt * tt;
    const float b2 = 3.f * mt * tt * tt;
    const float b3 = tt * tt * tt;
    const float* base = cp + (size_t)(((b * 2 + p) * 4 + s) * 4) * 2;
    qx[p][m] = b0 * base[0] + b1 * base[2] + b2 * base[4] + b3 * base[6];
    qy[p][m] = b0 * base[1] + b1 * base[3] + b2 * base[5] + b3 * base[7];
  }
  if (tid < 2) { sw[tid] = widths[b * 2 + tid]; sa[tid] = alphas[b * 2 + tid]; }
  __syncthreads();

  for (int pix = tid; pix < 28 * 28; pix += blockDim.x) {
    const int py = pix / 28, px = pix % 28;
    float accum = 0.f;
    #pragma unroll
    for (int ay = 0; ay < 2; ++ay) {
      #pragma unroll
      for (int ax = 0; ax < 2; ++ax) {
        const float sy = ((float)(py * 2 + ay) + 0.5f) * 0.5f;
        const float sx = ((float)(px * 2 + ax) + 0.5f) * 0.5f;
        float img = 1.0f;
        #pragma unroll
        for (int p = 0; p < 2; ++p) {
          float mind2 = 1e30f;
          int ncross = 0;
          for (int m = 0; m < 32; ++m) {
            const float x0 = qx[p][m], y0 = qy[p][m];
            const float dx = x0 - sx, dy = y0 - sy;
            mind2 = fminf(mind2, dx * dx + dy * dy);
            const int m1 = (m + 1) & 31;
            const float x1 = qx[p][m1], y1 = qy[p][m1];
            const bool band = (y0 > sy) != (y1 > sy);
            const float te  = (sy - y0) / (y1 - y0 + 1e-8f);
            const float xin = x0 + te * (x1 - x0);
            ncross += (band && (sx < xin)) ? 1 : 0;
          }
          const float dist   = sqrtf(mind2);
          const float stroke = fminf(fmaxf(sw[p] * 0.5f + 0.5f - dist, 0.f), 1.f);
          const float inside = (float)(ncross & 1);
          const float cov    = fmaxf(inside, stroke);
          img *= (1.0f - sa[p] * cov);
        }
        accum += img;
      }
    }
    rendered[(size_t)b * 784 + pix] = 1.0f - accum * 0.25f;
  }
}

// ---------------------------------------------------------------------------
extern "C" void kernel_launch(void* const* d_in, const int* in_sizes, int n_in,
                              void* d_out, int out_size, void* d_ws, size_t ws_size,
                              hipStream_t stream) {
  const float* x      = (const float*)d_in[0];
  const float* eps    = (const float*)d_in[1];
  const float* enc_w1 = (const float*)d_in[2];
  const float* enc_b1 = (const float*)d_in[3];
  const float* enc_w2 = (const float*)d_in[4];
  const float* enc_b2 = (const float*)d_in[5];
  const float* mu_w   = (const float*)d_in[6];
  const float* mu_b   = (const float*)d_in[7];
  const float* lv_w   = (const float*)d_in[8];
  const float* lv_b   = (const float*)d_in[9];
  const float* dec_w1 = (const float*)d_in[10];
  const float* dec_b1 = (const float*)d_in[11];
  const float* dec_w2 = (const float*)d_in[12];
  const float* dec_b2 = (const float*)d_in[13];
  const float* cp_w   = (const float*)d_in[14];
  const float* cp_b   = (const float*)d_in[15];
  const float* ref_w1 = (const float*)d_in[16];
  const float* ref_b1 = (const float*)d_in[17];
  const float* ref_w2 = (const float*)d_in[18];
  const float* ref_b2 = (const float*)d_in[19];
  const float* wd_w   = (const float*)d_in[20];
  const float* wd_b   = (const float*)d_in[21];
  const float* al_w   = (const float*)d_in[22];
  const float* al_b   = (const float*)d_in[23];

  float* out = (float*)d_out;
  float* ws  = (float*)d_ws;

  // enc1: [128,784]x[784,256], leaky_relu          (128 wave-tiles / 4 per blk)
  gemm_kernel<1><<<32, 128, 0, stream>>>(x, enc_w1, enc_b1, ws + WS_H1,
                                         128, 256, 784, 784, 256, 256);
  // enc2: [128,256]x[256,256], leaky_relu
  gemm_kernel<1><<<32, 128, 0, stream>>>(ws + WS_H1, enc_w2, enc_b2, ws + WS_H2,
                                         128, 256, 256, 256, 256, 256);
  // mu / logvar / z (fused reparameterization); z -> h_in[:, :64]
  latent_kernel<<<8, 128, 0, stream>>>(ws + WS_H2, mu_w, mu_b, lv_w, lv_b, eps,
                                       out + R_MU, out + R_LV, ws + WS_HIN);
  // dec1: z[128,64](ld 92) x [64,512], selu
  gemm_kernel<2><<<64, 128, 0, stream>>>(ws + WS_HIN, dec_w1, dec_b1, ws + WS_G1,
                                         128, 512, 64, 92, 512, 512);
  // dec2: [128,512]x[512,512], selu
  gemm_kernel<2><<<64, 128, 0, stream>>>(ws + WS_G1, dec_w2, dec_b2, ws + WS_H2D,
                                         128, 512, 512, 512, 512, 512);
  // cp head: tanh -> pts_norm into h_in[:, 64:92] (ldo 92)
  gemm_kernel<3><<<4, 128, 0, stream>>>(ws + WS_H2D, cp_w, cp_b, ws + WS_HIN + 64,
                                        128, 28, 512, 512, 28, 92);
  // widths / alphas heads (tiny)
  heads_kernel<<<2, 256, 0, stream>>>(ws + WS_H2D, wd_w, wd_b, al_w, al_b,
                                      out + R_WD, out + R_AL);
  // ref1: h_in[128,92] x [92,512], selu
  gemm_kernel<2><<<64, 128, 0, stream>>>(ws + WS_HIN, ref_w1, ref_b1, ws + WS_R1,
                                         128, 512, 92, 92, 512, 512);
  // ref2 + points + control-point scatter into d_out
  ref2_points_kernel<<<8, 128, 0, stream>>>(ws + WS_R1, ref_w2, ref_b2, out + R_CP);
  // rasterizer: 1 block per image
  render_kernel<<<128, 256, 0, stream>>>(out + R_CP, out + R_WD, out + R_AL,
                                         out + R_REND);
}